// NodeClassifier_17609365914133
// MI455X (gfx1250) — compile-verified
//
#include <hip/hip_runtime.h>

// NodeClassifier on MI455X (gfx1250):
//  - GCN aggregation via L2-resident global_atomic_add_f32 (x is 25.6MB << 192MB L2)
//  - All GEMMs on the fp32 tensor path V_WMMA_F32_16X16X4_F32
//  - B panels staged into LDS by the Tensor Data Mover (tensor_load_to_lds +
//    s_wait_tensorcnt); each wave register-blocks 4 M-tiles over one B fragment
//  - A fragments software-pipelined (double-buffered) so one wait covers a
//    batch of 4 loads and the k+4 loads overlap the k WMMAs
//  - BatchNorm statistics fused into GEMM epilogues (f32 atomics)

#define D_FEAT 128
#define BN_EPS 1e-5f
#define KMAX 512           // max K among GEMMs (ff2): LDS panel = KMAX*16*4B = 32KB
#define MT_PER_WAVE 4
#define WAVES 8

typedef float v2f __attribute__((ext_vector_type(2)));
typedef float v8f __attribute__((ext_vector_type(8)));
typedef unsigned int u32x4 __attribute__((ext_vector_type(4)));
typedef int i32x4 __attribute__((ext_vector_type(4)));
typedef int i32x8 __attribute__((ext_vector_type(8)));

#if __has_builtin(__builtin_amdgcn_tensor_load_to_lds) && \
    __has_builtin(__builtin_amdgcn_s_wait_tensorcnt)
#define USE_TDM 1
#else
#define USE_TDM 0
#endif

__device__ __forceinline__ void atomAddF(float* p, float v) {
#if defined(__HIP_DEVICE_COMPILE__)
  unsafeAtomicAdd(p, v);   // global_atomic_add_f32
#else
  atomicAdd(p, v);
#endif
}

__global__ void zero_kernel(float* __restrict__ p, long long n) {
  long long i = (long long)blockIdx.x * blockDim.x + threadIdx.x;
  long long stride = (long long)gridDim.x * blockDim.x;
  for (; i < n; i += stride) p[i] = 0.0f;
}

__global__ void deg_kernel(const int* __restrict__ dst, float* __restrict__ deg, int E) {
  int i = blockIdx.x * blockDim.x + threadIdx.x;
  if (i < E) atomAddF(&deg[dst[i]], 1.0f);
}

__global__ void invdeg_kernel(float* __restrict__ deg, int N) {
  int i = blockIdx.x * blockDim.x + threadIdx.x;
  if (i < N) deg[i] = 1.0f / fmaxf(deg[i], 1.0f);
}

// One edge per 32 lanes; each lane moves one float4 chunk of the 128-wide row.
__global__ void scatter_kernel(const float* __restrict__ x, const int* __restrict__ src,
                               const int* __restrict__ dst, float* __restrict__ agg, int E) {
  long long tid = (long long)blockIdx.x * blockDim.x + threadIdx.x;
  int e = (int)(tid >> 5);                 // D_FEAT/4 == 32 chunks per edge
  if (e >= E) return;
  int c = ((int)tid & 31) * 4;
  const float4 v = *(const float4*)(x + (size_t)src[e] * D_FEAT + c);
  float* p = agg + (size_t)dst[e] * D_FEAT + c;
  atomAddF(p + 0, v.x);
  atomAddF(p + 1, v.y);
  atomAddF(p + 2, v.z);
  atomAddF(p + 3, v.w);
}

// out[M,NC] = (rowscale? diag(rowscale):I) * A[M,K] @ B[K,NC] (+bias)(+resid)(ReLU?)
// Optionally accumulates per-column sum / sum^2 into stats for fused BatchNorm.
// Block = 8 waves: one 16-col panel per block, staged into LDS via TDM; each wave
// computes 4 16x16 M-tiles reusing the LDS B fragment. M,NC multiples of 16.
__global__ void __launch_bounds__(256)
gemm_wmma(const float* __restrict__ A, int lda,
          const float* __restrict__ rowscale,
          const float* __restrict__ B, int ldb,
          const float* __restrict__ bias,
          const float* __restrict__ resid,
          float* __restrict__ out,
          float* __restrict__ stats,
          int M, int NC, int K, int relu, int statD) {
  __shared__ float Bs[KMAX * 16];

  const int tilesN = NC >> 4;
  const int tilesM = M >> 4;
  const int colTile = blockIdx.x % tilesN;
  const int mBlock  = blockIdx.x / tilesN;
  const int col0 = colTile << 4;

  // ---- stage B[0:K, col0:col0+16] into LDS ----
#if USE_TDM
  if ((threadIdx.x >> 5) == 0) {           // one wave issues the DMA (EXEC ignored by TDM)
    unsigned long long ga = (unsigned long long)(uintptr_t)(B + col0);
    unsigned int ldsBase = (unsigned int)(uintptr_t)(void*)Bs;  // aperture low bits = LDS offset
    u32x4 g0;
    g0.x = 1u;                                              // count=1, user mode, no gather
    g0.y = ldsBase;                                         // lds_addr (bytes)
    g0.z = (unsigned int)(ga & 0xffffffffu);                // global_addr[31:0]
    g0.w = (unsigned int)((ga >> 32) & 0x01ffffffu) | (2u << 30);  // addr[56:32] | type=2
    i32x8 g1;
    g1[0] = (int)(2u << 16);                                // data_size = 4 bytes
    g1[1] = (int)(((unsigned)ldb & 0xffffu) << 16);         // tensor_dim0[15:0]
    g1[2] = (int)((((unsigned)ldb >> 16) & 0xffffu) |
                  (((unsigned)K & 0xffffu) << 16));         // dim0[31:16] | tensor_dim1[15:0]
    g1[3] = (int)(((((unsigned)K) >> 16) & 0xffffu) |
                  (16u << 16));                             // dim1[31:16] | tile_dim0=16
    g1[4] = (int)((unsigned)K & 0xffffu);                   // tile_dim1=K, tile_dim2=0
    g1[5] = (int)(unsigned)ldb;                             // tensor_dim0_stride[31:0]
    g1[6] = 0;
    g1[7] = 0;
    i32x4 z4 = {0, 0, 0, 0};
#if __clang_major__ >= 23
    i32x8 z8 = {0, 0, 0, 0, 0, 0, 0, 0};
    __builtin_amdgcn_tensor_load_to_lds(g0, g1, z4, z4, z8, 0);
#else
    __builtin_amdgcn_tensor_load_to_lds(g0, g1, z4, z4, 0);
#endif
    __builtin_amdgcn_s_wait_tensorcnt(0);
  }
#else
  for (int idx = threadIdx.x; idx < K * 16; idx += (int)blockDim.x)
    Bs[idx] = B[(size_t)(idx >> 4) * ldb + col0 + (idx & 15)];
#endif
  __syncthreads();

  const int wave = threadIdx.x >> 5;
  const int lane = threadIdx.x & 31;
  const int hi = lane >> 4;                 // 0: K slice {0,1}   1: K slice {2,3}
  const int l15 = lane & 15;

  const int mtBase = mBlock * (WAVES * MT_PER_WAVE) + wave * MT_PER_WAVE;
  if (mtBase >= tilesM) return;             // wave-uniform: EXEC stays all-ones for WMMA

  int tm[MT_PER_WAVE];
  const float* Arow[MT_PER_WAVE];
  float rs[MT_PER_WAVE];
#pragma unroll
  for (int mt = 0; mt < MT_PER_WAVE; ++mt) {
    int t = mtBase + mt;
    if (t > tilesM - 1) t = tilesM - 1;     // clamp: keep loads in-bounds, discard at store
    tm[mt] = t;
    const int row = (t << 4) + l15;
    Arow[mt] = A + (size_t)row * lda;
    rs[mt] = rowscale ? rowscale[row] : 1.0f;
  }

  v8f acc[MT_PER_WAVE];
#pragma unroll
  for (int mt = 0; mt < MT_PER_WAVE; ++mt) acc[mt] = (v8f){0, 0, 0, 0, 0, 0, 0, 0};

  const int h2 = hi * 2;

  // prologue: batch-load all 4 A fragments for k = 0 (one in-order wait covers all)
  v2f aReg[MT_PER_WAVE];
#pragma unroll
  for (int mt = 0; mt < MT_PER_WAVE; ++mt)
    aReg[mt] = *(const v2f*)(Arow[mt] + h2);

  for (int k = 0; k < K; k += 4) {
    // issue next iteration's A loads before consuming this iteration's fragments
    v2f aNext[MT_PER_WAVE];
    if (k + 4 < K) {
#pragma unroll
      for (int mt = 0; mt < MT_PER_WAVE; ++mt)
        aNext[mt] = *(const v2f*)(Arow[mt] + (k + 4) + h2);
    }
    if ((k & 63) == 0 && k + 64 < K)        // global_prefetch_b8 of upcoming A panel
      __builtin_prefetch(&Arow[0][k + 64], 0, 1);

    v2f b;
    b.x = Bs[(k + h2) * 16 + l15];
    b.y = Bs[(k + h2 + 1) * 16 + l15];

#pragma unroll
    for (int mt = 0; mt < MT_PER_WAVE; ++mt) {
      v2f a;
      a.x = aReg[mt].x * rs[mt];
      a.y = aReg[mt].y * rs[mt];
      acc[mt] = __builtin_amdgcn_wmma_f32_16x16x4_f32(false, a, false, b, (short)0,
                                                      acc[mt], false, false);
    }
#pragma unroll
    for (int mt = 0; mt < MT_PER_WAVE; ++mt) aReg[mt] = aNext[mt];
  }

  const int col = col0 + l15;
  const float bv = bias ? bias[col] : 0.0f;
#pragma unroll
  for (int mt = 0; mt < MT_PER_WAVE; ++mt) {
    if (mtBase + mt >= tilesM) break;       // clamped duplicates are never stored
    const int row0 = tm[mt] << 4;
    float s1 = 0.0f, s2 = 0.0f;
#pragma unroll
    for (int v = 0; v < 8; ++v) {
      const int row = row0 + v + hi * 8;    // C/D layout: VGPR v -> rows v (lanes<16) / v+8
      float val = acc[mt][v] + bv;
      if (resid) val += resid[(size_t)row * NC + col];
      if (relu) val = fmaxf(val, 0.0f);
      out[(size_t)row * NC + col] = val;
      s1 += val;
      s2 += val * val;
    }
    if (stats) {
      atomAddF(&stats[col], s1);
      atomAddF(&stats[col + statD], s2);
    }
  }
}

// x = (t - mean) * rsqrt(var + eps) * g + b   with mean/var from fused sums.
__global__ void bn_apply(const float* __restrict__ t, const float* __restrict__ stats,
                         const float* __restrict__ g, const float* __restrict__ b,
                         float* __restrict__ xout, long long total, int D, float invN) {
  long long i = (long long)blockIdx.x * blockDim.x + threadIdx.x;
  long long stride = (long long)gridDim.x * blockDim.x;
  for (; i < total; i += stride) {
    int d = (int)(i & (long long)(D - 1));   // D is a power of two (128)
    float m = stats[d] * invN;
    float var = stats[D + d] * invN - m * m;
    xout[i] = (t[i] - m) * rsqrtf(var + BN_EPS) * g[d] + b[d];
  }
}

extern "C" void kernel_launch(void* const* d_in, const int* in_sizes, int n_in,
                              void* d_out, int out_size, void* d_ws, size_t ws_size,
                              hipStream_t stream) {
  const float* nodes = (const float*)d_in[0];
  const float* W_gcn = (const float*)d_in[1];
  const float* b_gcn = (const float*)d_in[2];
  const float* bn1_g = (const float*)d_in[3];
  const float* bn1_b = (const float*)d_in[4];
  const float* bn2_g = (const float*)d_in[5];
  const float* bn2_b = (const float*)d_in[6];
  const float* ff_w1 = (const float*)d_in[7];
  const float* ff_b1 = (const float*)d_in[8];
  const float* ff_w2 = (const float*)d_in[9];
  const float* cls_w = (const float*)d_in[10];
  const float* cls_b = (const float*)d_in[11];
  const int* esrc = (const int*)d_in[12];
  const int* edst = (const int*)d_in[13];

  const int D = D_FEAT;
  const int N = in_sizes[0] / D;          // 50000
  const int E = in_sizes[12];             // 800000
  const int H = in_sizes[8] / 2;          // 512
  const int C = in_sizes[11];             // 16

  float* ws = (float*)d_ws;
  const size_t ND = (size_t)N * D;
  float* x     = ws;                      // [N,D] activations
  float* t     = x + ND;                  // [N,D] pre-BN buffer
  float* mid   = t + ND;                  // [N,H] FF hidden
  float* agg   = mid;                     // [N,D] aliases mid (disjoint lifetimes)
  float* invd  = mid + (size_t)N * H;     // [N]
  float* stats = invd + N;                // [2*D]

  const int tilesM = N / 16;
  const int mBlocks = (tilesM + WAVES * MT_PER_WAVE - 1) / (WAVES * MT_PER_WAVE);

  // in-degree -> 1/max(deg,1)
  zero_kernel<<<256, 256, 0, stream>>>(invd, N);
  deg_kernel<<<(E + 255) / 256, 256, 0, stream>>>(edst, invd, E);
  invdeg_kernel<<<(N + 255) / 256, 256, 0, stream>>>(invd, N);

  for (int i = 0; i < 2; ++i) {
    const float* xin = (i == 0) ? nodes : x;

    // agg = segment_sum(x[src], dst)
    zero_kernel<<<4096, 256, 0, stream>>>(agg, (long long)ND);
    const long long sThreads = (long long)E * 32;
    scatter_kernel<<<(int)((sThreads + 255) / 256), 256, 0, stream>>>(xin, esrc, edst, agg, E);

    // t = (agg * inv_deg) @ W_gcn[i] + b_gcn[i] + xin ; accumulate BN1 stats
    zero_kernel<<<1, 256, 0, stream>>>(stats, 2 * D);
    gemm_wmma<<<(D / 16) * mBlocks, 256, 0, stream>>>(
        agg, D, invd, W_gcn + (size_t)i * D * D, D,
        b_gcn + i * D, xin, t, stats, N, D, D, /*relu=*/0, D);
    bn_apply<<<4096, 256, 0, stream>>>(t, stats, bn1_g + i * D, bn1_b + i * D,
                                       x, (long long)ND, D, 1.0f / (float)N);

    // mid = relu(x @ ff_w1[i] + ff_b1[i])
    gemm_wmma<<<(H / 16) * mBlocks, 256, 0, stream>>>(
        x, D, nullptr, ff_w1 + (size_t)i * D * H, H,
        ff_b1 + i * H, nullptr, mid, nullptr, N, H, D, /*relu=*/1, 0);

    // t = mid @ ff_w2[i] + x ; accumulate BN2 stats
    zero_kernel<<<1, 256, 0, stream>>>(stats, 2 * D);
    gemm_wmma<<<(D / 16) * mBlocks, 256, 0, stream>>>(
        mid, H, nullptr, ff_w2 + (size_t)i * H * D, D,
        nullptr, x, t, stats, N, D, H, /*relu=*/0, D);
    bn_apply<<<4096, 256, 0, stream>>>(t, stats, bn2_g + i * D, bn2_b + i * D,
                                       x, (long long)ND, D, 1.0f / (float)N);
  }

  // logits = x @ cls_w + cls_b
  gemm_wmma<<<(C / 16) * mBlocks, 256, 0, stream>>>(
      x, D, nullptr, cls_w, C, cls_b, nullptr,
      (float*)d_out, nullptr, N, C, D, /*relu=*/0, 0);
}